// GNNGraphCustom_28080496181822
// MI455X (gfx1250) — compile-verified
//
#include <hip/hip_runtime.h>
#include <hip/hip_bf16.h>

typedef float v2f __attribute__((ext_vector_type(2)));
typedef float v8f __attribute__((ext_vector_type(8)));

#define DF 256
#define N_GRAPHS 128
#define N_CLASSES 10
#define BSTRIDE 260   // 256 + 4 pad floats -> conflict-free ds_load_b64

// ---------------- degree / norm ----------------
__global__ void k_deg_init(float* dinv, int N) {
  int i = blockIdx.x * blockDim.x + threadIdx.x;
  if (i < N) dinv[i] = 1.0f;              // self-loop contributes 1 to degree
}
__global__ void k_deg_count(const int* __restrict__ dst, float* deg, int E) {
  int e = blockIdx.x * blockDim.x + threadIdx.x;
  if (e < E) unsafeAtomicAdd(&deg[dst[e]], 1.0f);
}
__global__ void k_deg_rsqrt(float* dinv, int N) {
  int i = blockIdx.x * blockDim.x + threadIdx.x;
  if (i < N) dinv[i] = rsqrtf(dinv[i]);   // deg >= 1 always (self-loop)
}

// ---------------- per-layer elementwise: P = Y = in * dinv[row] ----------------
// (Y may alias `in`: per-element read-then-write is safe.)
__global__ void k_scale_dup(const float* __restrict__ in, const float* __restrict__ dinv,
                            float* __restrict__ P, float* __restrict__ Y, int NT) {
  int i = blockIdx.x * blockDim.x + threadIdx.x;
  if (i < NT) {
    float v = in[i] * dinv[i >> 8];
    P[i] = v;
    Y[i] = v;
  }
}

// ---------------- edge scatter: Y[dst] += P[src]  (one wave per edge) ----------------
// b128 gathers (2 per lane), then 8 hardware f32 atomics into L2-resident Y.
__global__ void k_edge_scatter(const int* __restrict__ src, const int* __restrict__ dst,
                               const float* __restrict__ P, float* __restrict__ Y, int E) {
  int wave = (blockIdx.x * blockDim.x + threadIdx.x) >> 5;
  int lane = threadIdx.x & 31;
  if (wave >= E) return;
  int s = src[wave];
  int d = dst[wave];
  const float4* ps = (const float4*)(P + (long long)s * DF);
  float*        yd = Y + (long long)d * DF;
  float4 v0 = ps[lane];
  float4 v1 = ps[lane + 32];
  int f0 = lane * 4;
  int f1 = (lane + 32) * 4;
  unsafeAtomicAdd(&yd[f0 + 0], v0.x);
  unsafeAtomicAdd(&yd[f0 + 1], v0.y);
  unsafeAtomicAdd(&yd[f0 + 2], v0.z);
  unsafeAtomicAdd(&yd[f0 + 3], v0.w);
  unsafeAtomicAdd(&yd[f1 + 0], v1.x);
  unsafeAtomicAdd(&yd[f1 + 1], v1.y);
  unsafeAtomicAdd(&yd[f1 + 2], v1.z);
  unsafeAtomicAdd(&yd[f1 + 3], v1.w);
}

// ---------------- 256x256 weight transpose (so B fragments load contiguously) ----------------
__global__ void k_transpose256(const float* __restrict__ W, float* __restrict__ Wt) {
  int i = blockIdx.x * blockDim.x + threadIdx.x;  // 65536
  int n = i >> 8, k = i & 255;
  Wt[i] = W[k * DF + n];                          // Wt[n][k] = W[k][n]
}

// ---------------- WMMA GEMM: OUT = relu( dinv[m] * (Y @ W) + b ) ----------------
// Grid: x = M-tile group (8 tiles/block, 1 per wave), y = N strip (64 cols).
// The block's 64x256 strip of Wt is staged once into LDS (padded stride 260:
// ds_load_b64 B-fragment reads hit all 64 banks exactly once).
// V_WMMA_F32_16X16X4_F32, K=256:
//   A (16x4): lanes 0-15 hold M=0..15,K={0,1}; lanes 16-31 same M, K={2,3}.
//   B (4x16): symmetric, lane holds a column of B = a row of Wt.
//   D (16x16): VGPR r, lanes 0-15 -> M=r, N=lane; lanes 16-31 -> M=8+r, N=lane-16.
__global__ void k_gemm_wmma(const float* __restrict__ Y, const float* __restrict__ Wt,
                            const float* __restrict__ dinv, const float* __restrict__ bias,
                            float* __restrict__ OUT, int Mtiles) {
  __shared__ float Bs[64 * BSTRIDE];
  const int ns = blockIdx.y;            // N strip (64 cols)

  // cooperative stage of Wt rows [ns*64, ns*64+64) -> LDS
  {
    const float4* s4 = (const float4*)(Wt + (long long)ns * 64 * DF);
    for (int idx = threadIdx.x; idx < 64 * 64; idx += blockDim.x) {
      int r = idx >> 6, k4 = idx & 63;
      *(float4*)(Bs + r * BSTRIDE + k4 * 4) = s4[r * 64 + k4];
    }
  }
  __syncthreads();

  const int lane = threadIdx.x & 31;
  const int mt   = blockIdx.x * 8 + (threadIdx.x >> 5);
  if (mt >= Mtiles) return;

  const int row = lane & 15;
  const int kh  = (lane >> 4) << 1;     // K sub-offset: 0 or 2
  const float* a_ptr = Y + (long long)(mt * 16 + row) * DF + kh;
  const float* b0p = Bs + (row     ) * BSTRIDE + kh;
  const float* b1p = Bs + (row + 16) * BSTRIDE + kh;
  const float* b2p = Bs + (row + 32) * BSTRIDE + kh;
  const float* b3p = Bs + (row + 48) * BSTRIDE + kh;

  v8f c0 = {0.f,0.f,0.f,0.f,0.f,0.f,0.f,0.f};
  v8f c1 = c0, c2 = c0, c3 = c0;

  for (int k = 0; k < DF; k += 4) {
    v2f a  = *(const v2f*)(a_ptr + k);
    v2f b0 = *(const v2f*)(b0p + k);
    v2f b1 = *(const v2f*)(b1p + k);
    v2f b2 = *(const v2f*)(b2p + k);
    v2f b3 = *(const v2f*)(b3p + k);
    c0 = __builtin_amdgcn_wmma_f32_16x16x4_f32(false, a, false, b0, (short)0, c0, false, false);
    c1 = __builtin_amdgcn_wmma_f32_16x16x4_f32(false, a, false, b1, (short)0, c1, false, false);
    c2 = __builtin_amdgcn_wmma_f32_16x16x4_f32(false, a, false, b2, (short)0, c2, false, false);
    c3 = __builtin_amdgcn_wmma_f32_16x16x4_f32(false, a, false, b3, (short)0, c3, false, false);
  }

  const int mbase = mt * 16 + (lane >> 4) * 8;
  const int col   = lane & 15;
  const float bb0 = bias[ns * 64 +  0 + col];
  const float bb1 = bias[ns * 64 + 16 + col];
  const float bb2 = bias[ns * 64 + 32 + col];
  const float bb3 = bias[ns * 64 + 48 + col];
  float* out = OUT + (long long)mbase * DF + ns * 64 + col;
#pragma unroll
  for (int r = 0; r < 8; ++r) {
    float di = dinv[mbase + r];
    float v0 = c0[r] * di + bb0;
    float v1 = c1[r] * di + bb1;
    float v2 = c2[r] * di + bb2;
    float v3 = c3[r] * di + bb3;
    out[(long long)r * DF +  0] = v0 > 0.f ? v0 : 0.f;
    out[(long long)r * DF + 16] = v1 > 0.f ? v1 : 0.f;
    out[(long long)r * DF + 32] = v2 > 0.f ? v2 : 0.f;
    out[(long long)r * DF + 48] = v3 > 0.f ? v3 : 0.f;
  }
}

// ---------------- global add pool ----------------
__global__ void k_zero(float* g, int n) {
  int i = blockIdx.x * blockDim.x + threadIdx.x;
  if (i < n) g[i] = 0.0f;
}
__global__ void k_pool(const float* __restrict__ h, const int* __restrict__ batch,
                       float* __restrict__ g, int NT) {
  int i = blockIdx.x * blockDim.x + threadIdx.x;
  if (i < NT) {
    int n = i >> 8, f = i & 255;
    unsafeAtomicAdd(&g[batch[n] * DF + f], h[i]);
  }
}

// ---------------- fc1 (128x256 @ 256x256, tiny) ----------------
__global__ void k_fc1(const float* __restrict__ g, const float* __restrict__ W,
                      const float* __restrict__ b, float* __restrict__ g2) {
  int i = blockIdx.x * blockDim.x + threadIdx.x;  // 128*256
  if (i < N_GRAPHS * DF) {
    int gi = i >> 8, j = i & 255;
    const float* row = g + gi * DF;
    float acc = b[j];
    for (int k = 0; k < DF; ++k) acc += row[k] * W[k * DF + j];
    g2[i] = acc > 0.f ? acc : 0.f;
  }
}

// ---------------- fc2 + log_softmax ----------------
__global__ void k_fc2_lsm(const float* __restrict__ g2, const float* __restrict__ W,
                          const float* __restrict__ b, float* __restrict__ out) {
  int gi = blockIdx.x * blockDim.x + threadIdx.x;
  if (gi >= N_GRAPHS) return;
  const float* row = g2 + gi * DF;
  float logit[N_CLASSES];
#pragma unroll
  for (int j = 0; j < N_CLASSES; ++j) {
    float acc = b[j];
    for (int k = 0; k < DF; ++k) acc += row[k] * W[k * N_CLASSES + j];
    logit[j] = acc;
  }
  float m = logit[0];
#pragma unroll
  for (int j = 1; j < N_CLASSES; ++j) m = logit[j] > m ? logit[j] : m;
  float s = 0.f;
#pragma unroll
  for (int j = 0; j < N_CLASSES; ++j) s += expf(logit[j] - m);
  float lse = logf(s);
#pragma unroll
  for (int j = 0; j < N_CLASSES; ++j) out[gi * N_CLASSES + j] = logit[j] - m - lse;
}

extern "C" void kernel_launch(void* const* d_in, const int* in_sizes, int n_in,
                              void* d_out, int out_size, void* d_ws, size_t ws_size,
                              hipStream_t stream) {
  const float* x     = (const float*)d_in[0];
  const int*   ei    = (const int*)d_in[1];
  const int*   batch = (const int*)d_in[2];
  const float* W0 = (const float*)d_in[3];  const float* b0 = (const float*)d_in[4];
  const float* W1 = (const float*)d_in[5];  const float* b1 = (const float*)d_in[6];
  const float* W2 = (const float*)d_in[7];  const float* b2 = (const float*)d_in[8];
  const float* fc1W = (const float*)d_in[9];  const float* fc1b = (const float*)d_in[10];
  const float* fc2W = (const float*)d_in[11]; const float* fc2b = (const float*)d_in[12];
  float* out = (float*)d_out;

  const int N  = in_sizes[0] / DF;     // 50000
  const int E  = in_sizes[1] / 2;      // 800000
  const int NT = N * DF;               // 12.8M
  const int Mtiles = N / 16;           // 3125 (N divisible by 16)
  const int* src = ei;
  const int* dst = ei + E;

  // workspace layout
  char* ws = (char*)d_ws;
  size_t featB = (size_t)NT * sizeof(float);          // 51.2 MB
  float* hA   = (float*)(ws);
  float* hB   = (float*)(ws + featB);
  float* dinv = (float*)(ws + 2 * featB);
  float* Wt   = (float*)(ws + 2 * featB + (((size_t)N * 4 + 255) & ~(size_t)255));
  float* g    = Wt + DF * DF;
  float* g2   = g + N_GRAPHS * DF;

  const int T = 256;
  // degrees -> dinv
  k_deg_init <<<(N + T - 1) / T, T, 0, stream>>>(dinv, N);
  k_deg_count<<<(E + T - 1) / T, T, 0, stream>>>(dst, dinv, E);
  k_deg_rsqrt<<<(N + T - 1) / T, T, 0, stream>>>(dinv, N);

  const dim3 gemmGrid((Mtiles + 7) / 8, 4);           // 8 M-tiles/block, 4 N strips
  const int scatBlocks = (E * 32 + T - 1) / T;
  const int ntBlocks   = (NT + T - 1) / T;

  // layer = aggregate-then-transform:  OUT = relu( dinv ⊙ Â0 (in ⊙ dinv) @ W + b )
  auto layer = [&](const float* in, const float* W, const float* bia,
                   float* P, float* Y, float* OUT) {
    k_transpose256<<<(DF * DF + T - 1) / T, T, 0, stream>>>(W, Wt);
    k_scale_dup   <<<ntBlocks, T, 0, stream>>>(in, dinv, P, Y, NT);
    k_edge_scatter<<<scatBlocks, T, 0, stream>>>(src, dst, P, Y, E);
    k_gemm_wmma   <<<gemmGrid, T, 0, stream>>>(Y, Wt, dinv, bia, OUT, Mtiles);
  };

  layer(x,  W0, b0, hA, hB, hA);   // L0: in=x,  P=hA, Y=hB, out=hA
  layer(hA, W1, b1, hB, hA, hB);   // L1: in=hA, P=hB, Y=hA(in-place ok), out=hB
  layer(hB, W2, b2, hA, hB, hA);   // L2: in=hB, P=hA, Y=hB(in-place ok), out=hA

  // pool + MLP head
  k_zero<<<(N_GRAPHS * DF + T - 1) / T, T, 0, stream>>>(g, N_GRAPHS * DF);
  k_pool<<<ntBlocks, T, 0, stream>>>(hA, batch, g, NT);
  k_fc1 <<<(N_GRAPHS * DF + T - 1) / T, T, 0, stream>>>(g, fc1W, fc1b, g2);
  k_fc2_lsm<<<(N_GRAPHS + 63) / 64, 64, 0, stream>>>(g2, fc2W, fc2b, out);
}